// MainEncoder_84550726189145
// MI455X (gfx1250) — compile-verified
//
#include <hip/hip_runtime.h>
#include <hip/hip_bf16.h>

// Problem constants (from the reference)
#define BQ 64    // batch
#define NQ 256   // block_num
#define LQ 64    // block_len (time steps)
#define HQ 256   // hidden
#define DQ 320   // H + L
#define G3 768   // 3*H

typedef float v2f __attribute__((ext_vector_type(2)));
typedef float v8f __attribute__((ext_vector_type(8)));

// ---------------------------------------------------------------------------
// Fast device math
// ---------------------------------------------------------------------------
__device__ __forceinline__ float fast_tanh(float x) {
#if __has_builtin(__builtin_amdgcn_tanhf)
    return __builtin_amdgcn_tanhf(x);     // v_tanh_f32 (gfx1250 TRANS op)
#else
    float e2 = __expf(2.0f * x);          // v_exp_f32
    return 1.0f - 2.0f / (e2 + 1.0f);     // safe at +/- inf
#endif
}

__device__ __forceinline__ float fast_sigmoid(float x) {
    return 1.0f / (1.0f + __expf(-x));
}

// ---------------------------------------------------------------------------
// Kernel 1: weight transposes for coalesced per-step reads.
//   W1hT [d<256][e<320]  = W1[e][d]       (h-part of W1, pre-transposed)
//   W_ihT[n<256][g<768]  = W_ih[g][n]
//   W_hhT[d<256][g<768]  = W_hh[g][d]
// ---------------------------------------------------------------------------
#define T_SEG1 (HQ * DQ)          // 81920
#define T_SEG2 (NQ * G3)          // 196608
#define T_TOTAL (T_SEG1 + 2 * T_SEG2)

__global__ __launch_bounds__(256) void transpose_kernel(
    const float* __restrict__ W1, const float* __restrict__ W_ih,
    const float* __restrict__ W_hh, float* __restrict__ W1hT,
    float* __restrict__ W_ihT, float* __restrict__ W_hhT) {
    int i = blockIdx.x * blockDim.x + threadIdx.x;
    if (i < T_SEG1) {
        int d = i / DQ, e = i % DQ;
        W1hT[i] = W1[e * DQ + d];
    } else if (i < T_SEG1 + T_SEG2) {
        int j = i - T_SEG1;
        int n = j / G3, g = j % G3;
        W_ihT[j] = W_ih[g * NQ + n];
    } else if (i < T_TOTAL) {
        int j = i - (T_SEG1 + T_SEG2);
        int d = j / G3, g = j % G3;
        W_hhT[j] = W_hh[g * HQ + d];
    }
}

// ---------------------------------------------------------------------------
// Kernel 2: xWT precompute via fp32 WMMA.
//   C[m,e] = b1[e] + sum_l X[m,l] * W1[e, H+l],  m = b*256+n  (M=16384, K=64, E=320)
// Stored transposed: xWT[b][e][n]  (so the scan kernel reads coalesced over n).
// One wave per 16x16 output tile; K consumed in 16 chunks of 4 via
// V_WMMA_F32_16X16X4_F32 (exact fp32 — matches the fp32 reference).
//
// f32 lane layouts (CDNA5 ISA 7.12.2):
//   A 16x4 : lanes 0-15 hold M=lane, VGPR0/1 = K0/K1; lanes 16-31 = K2/K3
//   B 4x16 : lanes 0-15 hold N=lane, VGPR0/1 = K0/K1; lanes 16-31 = K2/K3
//   D 16x16: VGPR v -> M = v + (lane<16 ? 0 : 8), N = lane&15
// ---------------------------------------------------------------------------
#define MT (BQ * NQ / 16)   // 1024 M-tiles
#define ET (DQ / 16)        // 20   E-tiles
#define XW_WAVES (MT * ET)  // 20480 waves -> 2560 blocks of 256 threads

__global__ __launch_bounds__(256) void xw_wmma_kernel(
    const float* __restrict__ X,    // [16384, 64] row-major (== input_data)
    const float* __restrict__ W1,   // [320, 320]
    const float* __restrict__ b1,   // [320]
    float* __restrict__ xWT) {      // [64][320][256]
    int wave = (blockIdx.x * blockDim.x + threadIdx.x) >> 5;
    int lane = threadIdx.x & 31;
    int tm = wave / ET;             // M tile
    int te = wave % ET;             // E tile

    int l16    = lane & 15;
    int khalf  = (lane >> 4) << 1;  // 0 or 2
    int mrow   = tm * 16 + l16;     // A row for this lane
    int ecol   = te * 16 + l16;     // B column (== D's N) for this lane

    const float* arow = X + mrow * LQ;            // X[m][l]
    const float* brow = W1 + ecol * DQ + HQ;      // W1[e][H + l]

    v8f c;
    float bias = b1[ecol];                        // same N for all 8 D regs
#pragma unroll
    for (int v = 0; v < 8; ++v) c[v] = bias;

#pragma unroll
    for (int k0 = 0; k0 < LQ; k0 += 4) {
        v2f a, bmat;
        a.x    = arow[k0 + khalf];
        a.y    = arow[k0 + khalf + 1];
        bmat.x = brow[k0 + khalf];
        bmat.y = brow[k0 + khalf + 1];
        c = __builtin_amdgcn_wmma_f32_16x16x4_f32(
                /*neg_a=*/false, a, /*neg_b=*/false, bmat,
                /*c_mod=*/(short)0, c, /*reuse_a=*/false, /*reuse_b=*/false);
    }

    // Scatter-store transposed: xWT[b][e][n]
    int mbase = tm * 16 + ((lane >> 4) << 3);     // +8 for upper half-wave
#pragma unroll
    for (int v = 0; v < 8; ++v) {
        int m = mbase + v;
        int b = m >> 8;
        int n = m & 255;
        xWT[(b * DQ + ecol) * NQ + n] = c[v];
    }
}

// ---------------------------------------------------------------------------
// Block-wide reductions (wave32 shuffles + LDS), red has 9 slots.
// ---------------------------------------------------------------------------
__device__ __forceinline__ float blk_max(float v, volatile float* red, int tid) {
#pragma unroll
    for (int o = 16; o > 0; o >>= 1) v = fmaxf(v, __shfl_xor(v, o, 32));
    if ((tid & 31) == 0) red[tid >> 5] = v;
    __syncthreads();
    if (tid == 0) {
        float m = red[0];
#pragma unroll
        for (int w = 1; w < 8; ++w) m = fmaxf(m, red[w]);
        red[8] = m;
    }
    __syncthreads();
    return red[8];
}

__device__ __forceinline__ float blk_sum(float v, volatile float* red, int tid) {
#pragma unroll
    for (int o = 16; o > 0; o >>= 1) v += __shfl_xor(v, o, 32);
    if ((tid & 31) == 0) red[tid >> 5] = v;
    __syncthreads();
    if (tid == 0) {
        float s = red[0];
#pragma unroll
        for (int w = 1; w < 8; ++w) s += red[w];
        red[8] = s;
    }
    __syncthreads();
    return red[8];
}

// ---------------------------------------------------------------------------
// Kernel 3: the full L=64 step scan. One block per batch row b (no cross-b
// coupling exists: softmax is over n within b; GRU recurrence is per b).
// h lives in LDS for the whole scan.
// ---------------------------------------------------------------------------
__global__ __launch_bounds__(256) void scan_kernel(
    const float* __restrict__ X,      // [B, N*L]
    const float* __restrict__ xWT,    // [B][D][N]
    const float* __restrict__ W1hT,   // [H][D]
    const float* __restrict__ w2,     // [D]
    const float* __restrict__ b2,     // [1]
    const float* __restrict__ W_ihT,  // [N][3H]
    const float* __restrict__ W_hhT,  // [H][3H]
    const float* __restrict__ b_ih,   // [3H]
    const float* __restrict__ b_hh,   // [3H]
    float* __restrict__ enc,          // [B][L][H]
    float* __restrict__ attn) {       // [B][L][N]
    __shared__ float s_h[HQ];
    __shared__ float s_hW[DQ];
    __shared__ float s_xt[NQ];
    __shared__ float s_w2[DQ];
    __shared__ float s_red[9];

    const int tid = threadIdx.x;
    const int b   = blockIdx.x;

    s_h[tid]  = 0.0f;
    s_w2[tid] = w2[tid];
    if (tid < 64) s_w2[HQ + tid] = w2[HQ + tid];
    const float b2v = b2[0];
    const float bi_r = b_ih[tid], bi_z = b_ih[tid + HQ], bi_n = b_ih[tid + 2 * HQ];
    const float bh_r = b_hh[tid], bh_z = b_hh[tid + HQ], bh_n = b_hh[tid + 2 * HQ];
    const float* xw = xWT + b * DQ * NQ;
    const float* xb = X + b * NQ * LQ;
    __syncthreads();

    for (int t = 0; t < LQ; ++t) {
        // ---- phase 1: hW[e] = sum_d h[d] * W1[e][d]  (coalesced via W1hT) ----
        float acc0 = 0.0f, acc1 = 0.0f;
        const int e2 = HQ + (tid & 63);           // e = 256..319, repeats every 64 threads
        for (int d = 0; d < HQ; ++d) {
            float hd = s_h[d];
            const float* wr = W1hT + d * DQ;
            acc0 = fmaf(hd, wr[tid], acc0);
            acc1 = fmaf(hd, wr[e2], acc1);
        }
        s_hW[tid] = acc0;
        if (tid < 64) s_hW[HQ + tid] = acc1;
        __syncthreads();

        // ---- phase 2: score e[b,n] = b2 + sum_e w2[e]*tanh(xWT + hW) ----
        float ev = b2v;
        for (int e = 0; e < DQ; ++e) {
            ev = fmaf(s_w2[e], fast_tanh(xw[e * NQ + tid] + s_hW[e]), ev);
        }

        // ---- phase 3: softmax over n (the 256 threads) ----
        float mx = blk_max(ev, s_red, tid);
        float p  = __expf(ev - mx);
        float sm = blk_sum(p, s_red, tid);
        float a  = p / sm;
        attn[(b * LQ + t) * NQ + tid] = a;
        s_xt[tid] = a * xb[tid * LQ + t];         // xt[n] = a[n]*x[b,n,t]
        __syncthreads();

        // ---- phase 4: GRU cell; thread j owns gates (r,z,n) of unit j ----
        float ir = bi_r, iz = bi_z, in_ = bi_n;
        float hr = bh_r, hz = bh_z, hn = bh_n;
        for (int n = 0; n < NQ; ++n) {
            float xv = s_xt[n];
            const float* wr = W_ihT + n * G3;
            ir  = fmaf(xv, wr[tid], ir);
            iz  = fmaf(xv, wr[tid + HQ], iz);
            in_ = fmaf(xv, wr[tid + 2 * HQ], in_);
        }
        for (int d = 0; d < HQ; ++d) {
            float hv = s_h[d];
            const float* wr = W_hhT + d * G3;
            hr = fmaf(hv, wr[tid], hr);
            hz = fmaf(hv, wr[tid + HQ], hz);
            hn = fmaf(hv, wr[tid + 2 * HQ], hn);
        }
        float r  = fast_sigmoid(ir + hr);
        float z  = fast_sigmoid(iz + hz);
        float nn = fast_tanh(in_ + r * hn);
        float hnew = (1.0f - z) * nn + z * s_h[tid];
        enc[(b * LQ + t) * HQ + tid] = hnew;
        __syncthreads();
        s_h[tid] = hnew;
        __syncthreads();
    }
}

// ---------------------------------------------------------------------------
// Launch
// ---------------------------------------------------------------------------
extern "C" void kernel_launch(void* const* d_in, const int* in_sizes, int n_in,
                              void* d_out, int out_size, void* d_ws, size_t ws_size,
                              hipStream_t stream) {
    (void)in_sizes; (void)n_in; (void)out_size; (void)ws_size;
    const float* X    = (const float*)d_in[0];
    const float* W1   = (const float*)d_in[1];
    const float* b1   = (const float*)d_in[2];
    const float* w2   = (const float*)d_in[3];
    const float* b2   = (const float*)d_in[4];
    const float* W_ih = (const float*)d_in[5];
    const float* W_hh = (const float*)d_in[6];
    const float* b_ih = (const float*)d_in[7];
    const float* b_hh = (const float*)d_in[8];

    float* enc  = (float*)d_out;                      // [B,L,H]
    float* attn = enc + (size_t)BQ * LQ * HQ;         // [B,L,N]

    float* ws    = (float*)d_ws;
    float* xWT   = ws;                                // 64*320*256 = 5,242,880 f
    float* W1hT  = xWT  + (size_t)BQ * DQ * NQ;       // 81,920 f
    float* W_ihT = W1hT + (size_t)HQ * DQ;            // 196,608 f
    float* W_hhT = W_ihT + (size_t)NQ * G3;           // 196,608 f  (~22.9 MB total)

    transpose_kernel<<<(T_TOTAL + 255) / 256, 256, 0, stream>>>(
        W1, W_ih, W_hh, W1hT, W_ihT, W_hhT);
    xw_wmma_kernel<<<XW_WAVES / 8, 256, 0, stream>>>(X, W1, b1, xWT);
    scan_kernel<<<BQ, 256, 0, stream>>>(X, xWT, W1hT, w2, b2,
                                        W_ihT, W_hhT, b_ih, b_hh, enc, attn);
}